// nodeselection_89730456748789
// MI455X (gfx1250) — compile-verified
//
#include <hip/hip_runtime.h>

#define Bdim 64
#define Ndim 8192
#define Tdim 64
#define Mdim 256
#define Kdim 32

typedef float v2f __attribute__((ext_vector_type(2)));
typedef float v8f __attribute__((ext_vector_type(8)));

#define NEG_BIG (-1.0e30f)

__global__ __launch_bounds__(128) void nodesel_fused_kernel(
    const float* __restrict__ nv1g, const float* __restrict__ nv2g,
    const float* __restrict__ gemb, float* __restrict__ out)
{
    __shared__ float emb[16 * 132];          // 16 rows x 128 dims, padded stride 132
    __shared__ float sc[4][16][17];          // per-wave score staging, padded
    __shared__ float topv[4][16][32];        // per-wave per-row top-32 values (desc)
    __shared__ int   topi[4][16][32];        // per-wave per-row top-32 indices
    __shared__ int   fidx[16][32];           // final merged indices per row

    const int tid   = threadIdx.x;
    const int bb    = blockIdx.x >> 4;       // batch
    const int mtile = blockIdx.x & 15;       // 16-row tile of M
    const int w     = tid >> 5;              // wave id (0..3), wave32
    const int lane  = tid & 31;
    const int col   = lane & 15;
    const int hi    = lane >> 4;

    const float* nv1 = nv1g + (size_t)bb * Ndim * Tdim;
    const float* nv2 = nv2g + (size_t)bb * Ndim * Tdim;

    // ---- stage embedding tile [16 x 128] into LDS (stride 132) ----
    {
        const float* src = gemb + (size_t)mtile * 16 * (2 * Tdim);
        for (int i = tid; i < 16 * 128; i += 128) {
            int r = i >> 7, c = i & 127;
            emb[r * 132 + c] = src[r * 128 + c];
        }
    }
    // ---- init per-wave top-k lists ----
    if (lane < 16) {
        for (int j = 0; j < 32; ++j) {
            topv[w][lane][j] = NEG_BIG;
            topi[w][lane][j] = 0;
        }
    }
    __syncthreads();

    // ---- A fragments (16x4 f32 per step, 32 steps over k=128) ----
    // lanes 0-15: M=lane, K = {4kk, 4kk+1}; lanes 16-31: M=lane-16, K = {4kk+2, 4kk+3}
    v2f a[32];
#pragma unroll
    for (int kk = 0; kk < 32; ++kk)
        a[kk] = *(const v2f*)&emb[col * 132 + 4 * kk + 2 * hi];

    // ---- stream n in 16-wide tiles; each wave owns a 2048-column quarter ----
    float thr = NEG_BIG;
    const int n0base = w * (Ndim / 4);
    for (int t = 0; t < (Ndim / 4) / 16; ++t) {
        const int n0 = n0base + t * 16;
        const float* r1 = nv1 + (size_t)(n0 + col) * Tdim + 2 * hi;
        const float* r2 = nv2 + (size_t)(n0 + col) * Tdim + 2 * hi;
        v8f acc = {0.f, 0.f, 0.f, 0.f, 0.f, 0.f, 0.f, 0.f};
#pragma unroll
        for (int kk = 0; kk < 16; ++kk) {
            v2f bf = *(const v2f*)(r1 + 4 * kk);   // dims [0,64) from nodevec1
            acc = __builtin_amdgcn_wmma_f32_16x16x4_f32(
                false, a[kk], false, bf, (short)0, acc, false, false);
        }
#pragma unroll
        for (int kk = 0; kk < 16; ++kk) {
            v2f bf = *(const v2f*)(r2 + 4 * kk);   // dims [64,128) from nodevec2
            acc = __builtin_amdgcn_wmma_f32_16x16x4_f32(
                false, a[16 + kk], false, bf, (short)0, acc, false, false);
        }
        // stage D: VGPR r holds row (hi*8 + r), column = lane&15
#pragma unroll
        for (int r = 0; r < 8; ++r)
            sc[w][hi * 8 + r][col] = acc[r];
        __builtin_amdgcn_wave_barrier();

        // lanes 0-15 each own one m-row: threshold-filtered sorted insertion
        if (lane < 16) {
            for (int c = 0; c < 16; ++c) {
                float v = sc[w][lane][c];
                if (v > thr) {
                    int j = 31;
                    while (j > 0 && topv[w][lane][j - 1] < v) {
                        topv[w][lane][j] = topv[w][lane][j - 1];
                        topi[w][lane][j] = topi[w][lane][j - 1];
                        --j;
                    }
                    topv[w][lane][j] = v;
                    topi[w][lane][j] = n0 + c;
                    thr = topv[w][lane][31];
                }
            }
        }
        __builtin_amdgcn_wave_barrier();
    }
    __syncthreads();

    // ---- output regions (tuple concatenated flat, ints promoted to f32) ----
    const size_t SEL = (size_t)Bdim * Mdim * Kdim * Tdim;   // 33,554,432
    const size_t BMK = (size_t)Bdim * Mdim * Kdim;          //    524,288
    float* out_sel1 = out;
    float* out_sel2 = out + SEL;
    float* out_bidx = out + 2 * SEL;
    float* out_idx  = out + 2 * SEL + BMK;

    // ---- merge 4 per-wave sorted lists -> final descending top-32 per row ----
    if (tid < 16) {
        const int row = tid;
        const int m = mtile * 16 + row;
        int p[4] = {0, 0, 0, 0};
        const size_t obase = ((size_t)bb * Mdim + m) * Kdim;
        for (int k = 0; k < Kdim; ++k) {
            float best = -__builtin_inff();
            int bw = 0;
#pragma unroll
            for (int ww = 0; ww < 4; ++ww) {
                if (p[ww] < 32) {
                    float v = topv[ww][row][p[ww]];
                    if (v > best) { best = v; bw = ww; }
                }
            }
            int nsel = topi[bw][row][p[bw]];
            p[bw]++;
            fidx[row][k] = nsel;
            out_idx[obase + k]  = (float)nsel;
            out_bidx[obase + k] = (float)bb;
        }
    }
    __syncthreads();

    // ---- gather sel1/sel2: 16 rows x 32 picks x 64 floats, float4 granularity ----
    for (int it = tid; it < 16 * Kdim * (Tdim / 4); it += 128) {
        const int r = it >> 9;           // row in tile
        const int k = (it >> 4) & 31;    // pick
        const int e = it & 15;           // float4 within 64-float row
        const int n = fidx[r][k];
        const int m = mtile * 16 + r;
        const size_t rowoff = ((size_t)bb * Mdim + m) * Kdim + k;
        const float4* s1 = (const float4*)(nv1 + (size_t)n * Tdim) + e;
        const float4* s2 = (const float4*)(nv2 + (size_t)n * Tdim) + e;
        ((float4*)out_sel1)[rowoff * 16 + e] = *s1;
        ((float4*)out_sel2)[rowoff * 16 + e] = *s2;
    }
}

extern "C" void kernel_launch(void* const* d_in, const int* in_sizes, int n_in,
                              void* d_out, int out_size, void* d_ws, size_t ws_size,
                              hipStream_t stream) {
    (void)in_sizes; (void)n_in; (void)out_size; (void)d_ws; (void)ws_size;
    const float* nv1  = (const float*)d_in[0];
    const float* nv2  = (const float*)d_in[1];
    const float* gemb = (const float*)d_in[2];
    float* out = (float*)d_out;
    dim3 grid(Bdim * (Mdim / 16));   // 1024 workgroups
    dim3 block(128);                 // 4 wave32 waves
    hipLaunchKernelGGL(nodesel_fused_kernel, grid, block, 0, stream,
                       nv1, nv2, gemb, out);
}